// AnchorExtractor_1245540515887
// MI455X (gfx1250) — compile-verified
//
#include <hip/hip_runtime.h>
#include <hip/hip_bf16.h>
#include <stdint.h>

// Problem constants (from the reference)
#define BSN 4
#define H0 64
#define W0 64
#define L0 (H0 * W0)     // 4096
#define ANCHOR_NUM 64

// ---------------------------------------------------------------------------
// Kernel 1: init scratch (conf_flat = 0, j_flat = 0, winner = -1)
// ---------------------------------------------------------------------------
__global__ void init_kernel(float* conf_flat, int* j_flat, int* winner) {
    int idx = blockIdx.x * blockDim.x + threadIdx.x;
    if (idx < BSN * L0) {
        conf_flat[idx] = 0.0f;
        j_flat[idx]    = 0;
        winner[idx]    = -1;
    }
}

// ---------------------------------------------------------------------------
// Kernel 2: duplicate resolution — largest match index m wins (deterministic)
// ---------------------------------------------------------------------------
__global__ void vote_kernel(const int* b_ids, const int* i_ids, int* winner, int M) {
    int m = blockIdx.x * blockDim.x + threadIdx.x;
    if (m < M) {
        int b = b_ids[m];
        int i = i_ids[m];
        atomicMax(&winner[b * L0 + i], m);
    }
}

// ---------------------------------------------------------------------------
// Kernel 3: scatter confidences / target indices for winning matches
// ---------------------------------------------------------------------------
__global__ void scatter_kernel(const float* mconf, const int* b_ids, const int* i_ids,
                               const int* j_ids, const int* winner,
                               float* conf_flat, int* j_flat, int M) {
    int m = blockIdx.x * blockDim.x + threadIdx.x;
    if (m < M) {
        int idx = b_ids[m] * L0 + i_ids[m];
        if (winner[idx] == m) {
            conf_flat[idx] = mconf[m];
            j_flat[idx]    = j_ids[m];
        }
    }
}

// ---------------------------------------------------------------------------
// Kernel 4: NMS mask. pad (0,1,0,1) with zeros + 2x2 stride-1 window max.
// mask = (conf > 0) && (conf == window_max)
// ---------------------------------------------------------------------------
__global__ void mask_kernel(const float* conf_flat, int* mask) {
    int idx = blockIdx.x * blockDim.x + threadIdx.x;
    if (idx >= BSN * L0) return;
    int l = idx % L0;
    int h = l / W0;
    int w = l % W0;
    float c  = conf_flat[idx];
    float r  = (w + 1 < W0) ? conf_flat[idx + 1]      : 0.0f;
    float d  = (h + 1 < H0) ? conf_flat[idx + W0]     : 0.0f;
    float rd = (w + 1 < W0 && h + 1 < H0) ? conf_flat[idx + W0 + 1] : 0.0f;
    float mx = fmaxf(fmaxf(c, r), fmaxf(d, rd));
    mask[idx] = (c > 0.0f && c == mx) ? 1 : 0;
}

// ---------------------------------------------------------------------------
// Kernel 5: per-batch ordered compaction of mask-true indices (raster order,
// matching stable argsort(!mask)), then emit anchors (zeros if count == 0).
// One 256-thread block per batch; 16 elements per thread; LDS scan.
// ---------------------------------------------------------------------------
__global__ void anchors_kernel(const int* mask, const int* j_flat, float* out) {
    __shared__ int s_cnt[256];
    __shared__ int s_list[L0];

    const int b = blockIdx.x;
    const int t = threadIdx.x;
    const int PER = L0 / 256;   // 16

    int localIdx[PER];
    int cnt = 0;
    int base = t * PER;
    #pragma unroll
    for (int k = 0; k < PER; ++k) {
        if (mask[b * L0 + base + k]) localIdx[cnt++] = base + k;
    }
    s_cnt[t] = cnt;
    __syncthreads();

    // Hillis-Steele inclusive scan over 256 counts
    for (int off = 1; off < 256; off <<= 1) {
        int add = (t >= off) ? s_cnt[t - off] : 0;
        __syncthreads();
        s_cnt[t] += add;
        __syncthreads();
    }

    int start = s_cnt[t] - cnt;    // exclusive prefix
    for (int k = 0; k < cnt; ++k) s_list[start + k] = localIdx[k];
    __syncthreads();

    int total = s_cnt[255];
    if (t < ANCHOR_NUM) {
        float y0 = 0.f, x0 = 0.f, y1 = 0.f, x1 = 0.f;
        if (total > 0) {
            int sel = s_list[t % total];
            int j   = j_flat[b * L0 + sel];
            y0 = (float)(sel / W0); x0 = (float)(sel % W0);
            y1 = (float)(j   / W0); x1 = (float)(j   % W0);
        }
        float* o = out + (size_t)(b * ANCHOR_NUM + t) * 4;
        o[0] = y0; o[1] = x0; o[2] = y1; o[3] = x1;
    }
}

// ---------------------------------------------------------------------------
// Kernel 6: bulk pass-through copy via CDNA5 async global<->LDS B128.
// The 24-bit instruction offset is added to BOTH the global and LDS addresses
// (ISA 15.18), so one address operand + offset:k*4096 drives 8 batched
// transfers: 4 KB in flight per wave per ASYNCcnt drain (vs 512 B unbatched).
// Non-temporal hints keep this 512 MB stream from evicting the 192 MB L2.
// ---------------------------------------------------------------------------
#define BATCH 8   // b128 chunks per thread per iteration

__global__ void copy_async_kernel(const float* __restrict__ src,
                                  float* __restrict__ dst,
                                  long nChunks /* 16-byte chunks */,
                                  long nTailFloats, long tailBase) {
    __shared__ __align__(16) char stage[BATCH * 256 * 16];   // 32 KB

    const int tid = threadIdx.x;
    // Generic LDS address carries the LDS byte offset in addr[31:0] (ISA 10.2).
    uint32_t lds = (uint32_t)(uintptr_t)(&stage[0]) + (uint32_t)tid * 16u;

    const long groupChunks = (long)BATCH * 256;              // 2048 chunks / group
    const long nGroups     = nChunks / groupChunks;

    for (long g = blockIdx.x; g < nGroups; g += gridDim.x) {
        long chunk = g * groupChunks + tid;     // k-th batch element at +k*256
        const float* s = src + chunk * 4;
        float*       d = dst + chunk * 4;
        asm volatile(
            "global_load_async_to_lds_b128 %0, %1, off th:TH_LOAD_NT\n\t"
            "global_load_async_to_lds_b128 %0, %1, off offset:4096 th:TH_LOAD_NT\n\t"
            "global_load_async_to_lds_b128 %0, %1, off offset:8192 th:TH_LOAD_NT\n\t"
            "global_load_async_to_lds_b128 %0, %1, off offset:12288 th:TH_LOAD_NT\n\t"
            "global_load_async_to_lds_b128 %0, %1, off offset:16384 th:TH_LOAD_NT\n\t"
            "global_load_async_to_lds_b128 %0, %1, off offset:20480 th:TH_LOAD_NT\n\t"
            "global_load_async_to_lds_b128 %0, %1, off offset:24576 th:TH_LOAD_NT\n\t"
            "global_load_async_to_lds_b128 %0, %1, off offset:28672 th:TH_LOAD_NT\n\t"
            "s_wait_asynccnt 0\n\t"
            "global_store_async_from_lds_b128 %2, %0, off th:TH_STORE_NT\n\t"
            "global_store_async_from_lds_b128 %2, %0, off offset:4096 th:TH_STORE_NT\n\t"
            "global_store_async_from_lds_b128 %2, %0, off offset:8192 th:TH_STORE_NT\n\t"
            "global_store_async_from_lds_b128 %2, %0, off offset:12288 th:TH_STORE_NT\n\t"
            "global_store_async_from_lds_b128 %2, %0, off offset:16384 th:TH_STORE_NT\n\t"
            "global_store_async_from_lds_b128 %2, %0, off offset:20480 th:TH_STORE_NT\n\t"
            "global_store_async_from_lds_b128 %2, %0, off offset:24576 th:TH_STORE_NT\n\t"
            "global_store_async_from_lds_b128 %2, %0, off offset:28672 th:TH_STORE_NT\n\t"
            "s_wait_asynccnt 0"
            :
            : "v"(lds), "v"(s), "v"(d)
            : "memory");
    }

    // remainder chunks (single-b128 path)
    long done = nGroups * groupChunks;
    for (long i = done + (long)blockIdx.x * blockDim.x + tid; i < nChunks;
         i += (long)gridDim.x * blockDim.x) {
        const float* s = src + i * 4;
        float*       d = dst + i * 4;
        asm volatile(
            "global_load_async_to_lds_b128 %0, %1, off th:TH_LOAD_NT\n\t"
            "s_wait_asynccnt 0\n\t"
            "global_store_async_from_lds_b128 %2, %0, off th:TH_STORE_NT\n\t"
            "s_wait_asynccnt 0"
            :
            : "v"(lds), "v"(s), "v"(d)
            : "memory");
    }

    // scalar tail (only if total float count not divisible by 4)
    if (blockIdx.x == 0) {
        long tt = threadIdx.x;
        if (tt < nTailFloats) dst[tailBase + tt] = src[tailBase + tt];
    }
}

// ---------------------------------------------------------------------------
// Launch
// ---------------------------------------------------------------------------
extern "C" void kernel_launch(void* const* d_in, const int* in_sizes, int n_in,
                              void* d_out, int out_size, void* d_ws, size_t ws_size,
                              hipStream_t stream) {
    const float* conf_matrix = (const float*)d_in[0];
    const float* mconf       = (const float*)d_in[1];
    const int*   b_ids       = (const int*)d_in[2];
    const int*   i_ids       = (const int*)d_in[3];
    const int*   j_ids       = (const int*)d_in[4];

    const int M     = in_sizes[1];
    const long nCM  = (long)in_sizes[0];      // conf_matrix element count

    float* out_anchors = (float*)d_out;                           // [4,64,2,2]
    float* out_cm      = (float*)d_out + BSN * ANCHOR_NUM * 4;    // conf_matrix copy

    // scratch layout in d_ws
    float* conf_flat = (float*)d_ws;                  // BSN*L0 floats
    int*   j_flat    = (int*)(conf_flat + BSN * L0);  // BSN*L0 ints
    int*   winner    = j_flat + BSN * L0;             // BSN*L0 ints
    int*   mask      = winner + BSN * L0;             // BSN*L0 ints

    const int T = 256;
    const int gridGrid  = (BSN * L0 + T - 1) / T;     // 64 blocks
    const int gridMatch = (M + T - 1) / T;

    init_kernel<<<gridGrid, T, 0, stream>>>(conf_flat, j_flat, winner);
    vote_kernel<<<gridMatch, T, 0, stream>>>(b_ids, i_ids, winner, M);
    scatter_kernel<<<gridMatch, T, 0, stream>>>(mconf, b_ids, i_ids, j_ids,
                                                winner, conf_flat, j_flat, M);
    mask_kernel<<<gridGrid, T, 0, stream>>>(conf_flat, mask);
    anchors_kernel<<<BSN, T, 0, stream>>>(mask, j_flat, out_anchors);

    // bulk pass-through copy (~512 MB HBM traffic -> ~22 us at 23.3 TB/s)
    long nChunks    = nCM / 4;          // 16-byte chunks
    long tailBase   = nChunks * 4;
    long nTail      = nCM - tailBase;
    copy_async_kernel<<<2048, T, 0, stream>>>(conf_matrix, out_cm,
                                              nChunks, nTail, tailBase);
}